// NA_CPG_Matrix_Form_73701638799739
// MI455X (gfx1250) — compile-verified
//
#include <hip/hip_runtime.h>
#include <hip/hip_bf16.h>

typedef __attribute__((ext_vector_type(2))) float v2f;
typedef __attribute__((ext_vector_type(8))) float v8f;

#define ALPHA_C  1.0f
#define H_COUP_C 0.1f
#define DT_C     0.01f
#define N_OSC    4096
#define ROWS_PB  16
#define THREADS  256
#define NWAVES   (THREADS / 32)

__global__ __launch_bounds__(THREADS)
void cpg_matvec_wmma(const float* __restrict__ x_old,
                     const float* __restrict__ y_old,
                     const float* __restrict__ x_old_dot,
                     const float* __restrict__ omega,
                     const float* __restrict__ A,
                     const float* __restrict__ b,
                     const float* __restrict__ h_a,
                     const float* __restrict__ pd,
                     const float* __restrict__ y_state,
                     float* __restrict__ out)
{
    // Concat-layout state vector, split into even/odd interleaved components:
    // ux[j] = sv[2j], uy[j] = sv[2j+1] where sv = [x_old ; y_old - b]
    __shared__ float s_ux[N_OSC];
    __shared__ float s_uy[N_OSC];
    __shared__ float s_acc[ROWS_PB];

    const int tid = threadIdx.x;

    for (int j = tid; j < N_OSC; j += THREADS) {
        const int p0 = 2 * j, p1 = 2 * j + 1;
        s_ux[j] = (p0 < N_OSC) ? x_old[p0] : (y_old[p0 - N_OSC] - b[p0 - N_OSC]);
        s_uy[j] = (p1 < N_OSC) ? x_old[p1] : (y_old[p1 - N_OSC] - b[p1 - N_OSC]);
    }
    if (tid < ROWS_PB) s_acc[tid] = 0.0f;
    __syncthreads();

    const int lane = tid & 31;
    const int wave = tid >> 5;
    const int rloc = lane & 15;       // A-matrix: lane L -> row L & 15
    const int half = lane >> 4;       // 0: K={0,1}, 1: K={2,3}
    const int rowG = blockIdx.x * ROWS_PB + rloc;
    const float* __restrict__ pdrow = pd + (size_t)rowG * N_OSC;

    v2f bones;  bones.x = 1.0f;  bones.y = 1.0f;   // all-ones B => D[m][n] = rowsum(A) + C
    v8f acc = {};

    // 8 waves stride the 4096 columns in groups of 4 (one 16x4 A tile each).
    for (int base = wave * 4; base < N_OSC; base += NWAVES * 4) {
        __builtin_prefetch((const void*)(pdrow + base + NWAVES * 16), 0, 0);

        const int k0 = base + half * 2;
        const int k1 = k0 + 1;
        const float p0 = pdrow[k0];
        const float p1 = pdrow[k1];

        float s0, c0, s1, c1;
        __sincosf(p0, &s0, &c0);
        __sincosf(p1, &s1, &c1);

        // pd diagonal forced to zero in the reference: sin=0, cos=1
        const bool d0 = (k0 == rowG);
        const bool d1 = (k1 == rowG);
        s0 = d0 ? 0.0f : s0;   c0 = d0 ? 1.0f : c0;
        s1 = d1 ? 0.0f : s1;   c1 = d1 ? 1.0f : c1;

        v2f a;
        a.x = fmaf(s0, s_ux[k0], c0 * s_uy[k0]);
        a.y = fmaf(s1, s_ux[k1], c1 * s_uy[k1]);

        // f32 WMMA as a full-precision row-reduction: acc[m][*] += sum_k a[m][k]
        acc = __builtin_amdgcn_wmma_f32_16x16x4_f32(false, a, false, bones,
                                                    (short)0, acc, false, false);
    }

    // D layout (16x16 f32): VGPR r holds M=r on lanes 0-15, M=r+8 on lanes 16-31.
    // B was all-ones so any N column works; read N=0 from lanes 0 and 16.
    if (lane == 0) {
        #pragma unroll
        for (int r = 0; r < 8; ++r) atomicAdd(&s_acc[r], acc[r]);
    } else if (lane == 16) {
        #pragma unroll
        for (int r = 0; r < 8; ++r) atomicAdd(&s_acc[8 + r], acc[r]);
    }
    __syncthreads();

    if (tid < ROWS_PB) {
        const int i  = blockIdx.x * ROWS_PB + tid;
        const float xo = x_old[i];
        const float yo = y_old[i];
        const float bi = b[i];
        const float ym = yo - bi;
        const float r_sq = xo * xo + ym * ym;
        const float kd   = ALPHA_C * (1.0f - r_sq);
        const float zeta = 1.0f - h_a[i] * (xo / (fabsf(x_old_dot[i]) + 1e-9f));
        const float wz   = omega[i] / zeta;

        // dot[2i+1] = wz*u[2i] + kd*u[2i+1] + H_COUP * rowsum
        float dotv = wz * s_ux[i] + kd * s_uy[i] + H_COUP_C * s_acc[tid];
        const float mx = 2.0f * omega[i];
        dotv = fminf(fmaxf(dotv, -mx), mx);

        out[i] = A[i] * (y_state[i] + (dotv + bi) * DT_C);
    }
}

extern "C" void kernel_launch(void* const* d_in, const int* in_sizes, int n_in,
                              void* d_out, int out_size, void* d_ws, size_t ws_size,
                              hipStream_t stream) {
    (void)n_in; (void)d_ws; (void)ws_size;
    const float* x_old     = (const float*)d_in[0];
    const float* y_old     = (const float*)d_in[1];
    const float* x_old_dot = (const float*)d_in[2];
    const float* omega     = (const float*)d_in[3];
    const float* A         = (const float*)d_in[4];
    const float* b         = (const float*)d_in[5];
    const float* h_a       = (const float*)d_in[6];
    const float* pd        = (const float*)d_in[7];
    // d_in[8] = x_state: unused by the reference output
    const float* y_state   = (const float*)d_in[9];
    float* out = (float*)d_out;

    const int n = in_sizes[0];          // 4096
    const int grid = (n + ROWS_PB - 1) / ROWS_PB;
    cpg_matvec_wmma<<<grid, THREADS, 0, stream>>>(x_old, y_old, x_old_dot, omega,
                                                  A, b, h_a, pd, y_state, out);
    (void)out_size;
}